// SelfAttention_16346645528810
// MI455X (gfx1250) — compile-verified
//
#include <hip/hip_runtime.h>

typedef __attribute__((ext_vector_type(2))) float v2f;
typedef __attribute__((ext_vector_type(8))) float v8f;
typedef __attribute__((ext_vector_type(4))) float f4;
typedef __attribute__((ext_vector_type(4))) unsigned int v4u;
typedef __attribute__((ext_vector_type(8))) int v8i;
typedef __attribute__((ext_vector_type(4))) int v4i;

#define B_    2
#define S_    1024
#define POS_  1024
#define L_    2048
#define H_    16
#define D_    128
#define HID_  2048
#define BS_   (B_ * S_)

// ---------------------------------------------------------------------------
// Tensor Data Mover: issue a 2D f32 tile load (global -> LDS), one per wave.
// D# packing per CDNA5 ISA ch.8: group0 = {count, lds_addr, global_addr, type},
// group1 = {data_size/pad, tensor dims == tile dims, dim0 stride}.
// d0 = elements per row, d1 = rows, stride = row stride in elements.
// pen/pint/pamt: LDS pad enable / interval code (6=128 DW, 4=32 DW) / amount
// code (3 = 4 DWORDs) -> gives padded LDS rows with no extra instructions.
// ---------------------------------------------------------------------------
__device__ __forceinline__ void tdm_load_f32(unsigned lds, const float* g,
                                             unsigned d0, unsigned d1,
                                             unsigned stride, unsigned pen,
                                             unsigned pint, unsigned pamt)
{
  unsigned long long ga = (unsigned long long)(size_t)g;
  v4u g0;
  g0[0] = 1u;                                      // count=1, user descriptor
  g0[1] = lds;                                     // lds_addr (bytes)
  g0[2] = (unsigned)ga;                            // global_addr[31:0]
  g0[3] = ((unsigned)(ga >> 32) & 0x01FFFFFFu)     // global_addr[56:32]
          | (2u << 30);                            // type = 2 ("image")
  v8i g1;
  g1[0] = (int)((2u << 16) | (pen << 20) | (pint << 22) | (pamt << 25)); // 4B
  g1[1] = (int)((d0 & 0xFFFFu) << 16);             // tensor_dim0[15:0]
  g1[2] = (int)(((d0 >> 16) & 0xFFFFu) | ((d1 & 0xFFFFu) << 16)); // dim0 hi, dim1 lo
  g1[3] = (int)(((d1 >> 16) & 0xFFFFu) | ((d0 & 0xFFFFu) << 16)); // dim1 hi, tile_dim0
  g1[4] = (int)(d1 & 0xFFFFu);                     // tile_dim1 (tile_dim2 = 0)
  g1[5] = (int)stride;                             // tensor_dim0_stride[31:0]
  g1[6] = 0;
  g1[7] = 0;
  v4i z4 = {0, 0, 0, 0};
#if defined(__clang_major__) && (__clang_major__ >= 23)
  v8i z8 = {0, 0, 0, 0, 0, 0, 0, 0};
  __builtin_amdgcn_tensor_load_to_lds(g0, g1, z4, z4, z8, 0);
#else
  __builtin_amdgcn_tensor_load_to_lds(g0, g1, z4, z4, 0);
#endif
}

__device__ __forceinline__ unsigned lds_off(const void* p)
{
  return (unsigned)(size_t)p;
}

// ---------------------------------------------------------------------------
// Y[M,N] = A[M,K] @ W[N,K]^T + bias[N]   (M=N=K=2048)
// 64x64 output tile per 256-thread block; 8 waves in 4x2 layout, two 16x16
// v_wmma_f32_16x16x4_f32 tiles per wave. K tiles are TDM-DMA'd into a
// double-buffered LDS (rows padded 32->36 DWORDs by the TDM pad feature).
// ---------------------------------------------------------------------------
__global__ __launch_bounds__(256) void gemm_bias_wmma(
    const float* __restrict__ A, const float* __restrict__ W,
    const float* __restrict__ bias, float* __restrict__ Y)
{
  const int K = HID_, N = HID_;
  __shared__ __align__(16) float As[2][64][36];
  __shared__ __align__(16) float Ws[2][64][36];

  const int tid  = threadIdx.x;
  const int lane = tid & 31;
  const int w    = tid >> 5;       // 0..7
  const int wr   = w & 3;          // wave row 0..3  (16 rows each)
  const int wc   = w >> 2;         // wave col 0..1  (32 cols each)
  const int lr   = lane & 15;
  const bool hi  = lane >= 16;
  const int m0   = blockIdx.y * 64;
  const int n0   = blockIdx.x * 64;

  const float* Ag = A + (size_t)m0 * K;
  const float* Wg = W + (size_t)n0 * K;

  v8f acc0 = {}; v8f acc1 = {};

  const int NKT = K / 32;          // 64
  if (w == 0) {                    // prologue DMA: tile 0 -> buffer 0
    tdm_load_f32(lds_off(&As[0][0][0]), Ag, 32, 64, K, 1, 4, 3);
    tdm_load_f32(lds_off(&Ws[0][0][0]), Wg, 32, 64, K, 1, 4, 3);
  }

  for (int kt = 0; kt < NKT; ++kt) {
    const int cur = kt & 1;
    if (w == 0) {
      if (kt + 1 < NKT) {
        tdm_load_f32(lds_off(&As[cur ^ 1][0][0]), Ag + (kt + 1) * 32, 32, 64, K, 1, 4, 3);
        tdm_load_f32(lds_off(&Ws[cur ^ 1][0][0]), Wg + (kt + 1) * 32, 32, 64, K, 1, 4, 3);
        __builtin_amdgcn_s_wait_tensorcnt(2);   // next tile in flight, current done
      } else {
        __builtin_amdgcn_s_wait_tensorcnt(0);   // drain on last iteration
      }
    }
    __syncthreads();
#pragma unroll
    for (int ks = 0; ks < 8; ++ks) {
      const int ca = ks * 4 + (hi ? 2 : 0);
      v2f a, b0, b1;
      // A frag (16x4): lanes 0-15 rows, VGPR0/1 = K{0,1} (K{2,3} hi half)
      a.x  = As[cur][16 * wr + lr][ca];          a.y  = As[cur][16 * wr + lr][ca + 1];
      // B frag (4x16): B[k][n] = W[n][k]
      b0.x = Ws[cur][32 * wc + lr][ca];          b0.y = Ws[cur][32 * wc + lr][ca + 1];
      b1.x = Ws[cur][32 * wc + 16 + lr][ca];     b1.y = Ws[cur][32 * wc + 16 + lr][ca + 1];
      acc0 = __builtin_amdgcn_wmma_f32_16x16x4_f32(false, a, false, b0, (short)0, acc0, false, false);
      acc1 = __builtin_amdgcn_wmma_f32_16x16x4_f32(false, a, false, b1, (short)0, acc1, false, false);
    }
    __syncthreads();
  }

  const int col0 = n0 + 32 * wc + lr;
  const int col1 = col0 + 16;
  const float bv0 = bias[col0], bv1 = bias[col1];
#pragma unroll
  for (int v = 0; v < 8; ++v) {            // C/D layout: VGPR v = row v (+8 hi half)
    int row = m0 + 16 * wr + v + (hi ? 8 : 0);
    Y[(size_t)row * N + col0] = acc0[v] + bv0;
    Y[(size_t)row * N + col1] = acc1[v] + bv1;
  }
}

// ---------------------------------------------------------------------------
// RoPE on q,k (rot_dim == D == 128); scatter q -> (B,H,S,D),
// k,v -> (B,H,L,D) rows POS..L-1; also emit k_out/v_out (B,S,H,D).
// ---------------------------------------------------------------------------
__global__ void rope_scatter(
    const float* __restrict__ qr, const float* __restrict__ kr,
    const float* __restrict__ vr, const float* __restrict__ cosp,
    const float* __restrict__ sinp, float* __restrict__ qws,
    float* __restrict__ kfull, float* __restrict__ vfull,
    float* __restrict__ kout, float* __restrict__ vout)
{
  int t = blockIdx.x * blockDim.x + threadIdx.x;   // B*S*H*64 threads
  int d = t & 63;
  int h = (t >> 6) & 15;
  int s = (t >> 10) & 1023;
  int b = t >> 20;
  size_t base = ((size_t)(b * S_ + s)) * HID_ + h * D_;
  float c  = cosp[(POS_ + s) * 64 + d];
  float sn = sinp[(POS_ + s) * 64 + d];
  float q1 = qr[base + d], q2 = qr[base + d + 64];
  float k1 = kr[base + d], k2 = kr[base + d + 64];
  float v1 = vr[base + d], v2 = vr[base + d + 64];
  float qre = c * q1 - sn * q2, qim = sn * q1 + c * q2;
  float kre = c * k1 - sn * k2, kim = sn * k1 + c * k2;
  size_t qd = ((size_t)(b * H_ + h) * S_ + s) * D_ + d;
  qws[qd] = qre;  qws[qd + 64] = qim;
  size_t kf = ((size_t)(b * H_ + h) * L_ + (POS_ + s)) * D_ + d;
  kfull[kf] = kre;  kfull[kf + 64] = kim;
  vfull[kf] = v1;   vfull[kf + 64] = v2;
  size_t ko = ((size_t)(b * S_ + s) * H_ + h) * D_ + d;
  kout[ko] = kre;   kout[ko + 64] = kim;
  vout[ko] = v1;    vout[ko + 64] = v2;
}

// k_cache/v_cache (B,2,POS,H,D) layer 0  ->  (B,H,L,D) rows 0..POS-1
__global__ void cache_copy(const float* __restrict__ kc,
                           const float* __restrict__ vc,
                           float* __restrict__ kfull, float* __restrict__ vfull)
{
  int t = blockIdx.x * blockDim.x + threadIdx.x;   // B*H*POS*D threads
  int d = t & 127;
  int p = (t >> 7) & 1023;
  int h = (t >> 17) & 15;
  int b = t >> 21;
  size_t src = (((size_t)(b * 2) * POS_ + p) * H_ + h) * D_ + d;   // layer_id = 0
  size_t dst = ((size_t)(b * H_ + h) * L_ + p) * D_ + d;
  kfull[dst] = kc[src];
  vfull[dst] = vc[src];
}

// ---------------------------------------------------------------------------
// Flash attention, fp32 WMMA. Block = 4 waves, each wave owns 16 q rows kept
// in registers as A-fragments. Online softmax over L in 16-column chunks.
// K/V chunks are TDM-DMA'd into double-buffered LDS (K rows padded to 132
// DWORD-stride via TDM pad: interval 128 DW, amount 4 DW).
// ---------------------------------------------------------------------------
__global__ __launch_bounds__(128) void flash_attn(
    const float* __restrict__ qws, const float* __restrict__ kfull,
    const float* __restrict__ vfull, const float* __restrict__ mask,
    float* __restrict__ attn)
{
  __shared__ __align__(16) float Ks[2][16][132];
  __shared__ __align__(16) float Vs[2][16][128];
  __shared__ float Ps[4][16][17];

  const int tid  = threadIdx.x;
  const int lane = tid & 31;
  const int w    = tid >> 5;        // wave 0..3
  const int lr   = lane & 15;
  const bool hi  = lane >= 16;
  const int bh   = blockIdx.x;      // 0..31
  const int b    = bh >> 4, h = bh & 15;
  const int srow = blockIdx.y * 64 + w * 16;   // this wave's first q row

  // Q held as 32 A-fragments (K = D = 128 in steps of 4)
  v2f qa[32];
  const float* qg = qws + ((size_t)bh * S_ + srow) * D_;
#pragma unroll
  for (int ks = 0; ks < 32; ++ks) {
    const float* p = qg + (size_t)lr * D_ + ks * 4 + (hi ? 2 : 0);
    qa[ks].x = p[0]; qa[ks].y = p[1];
  }

  v8f acc[8];
#pragma unroll
  for (int i = 0; i < 8; ++i) acc[i] = (v8f){};
  float m_run[8], l_run[8];
#pragma unroll
  for (int v = 0; v < 8; ++v) { m_run[v] = -1e30f; l_run[v] = 0.f; }

  const float invscale = 0.08838834764831845f;  // 1/sqrt(128)
  const float* kbase = kfull + (size_t)bh * L_ * D_;
  const float* vbase = vfull + (size_t)bh * L_ * D_;
  const float* mbase = mask + (size_t)b * L_ * L_ + (size_t)POS_ * L_;

  const int NC = L_ / 16;           // 128 chunks
  if (w == 0) {                     // prologue DMA: chunk 0 -> buffer 0
    tdm_load_f32(lds_off(&Ks[0][0][0]), kbase, 16 * D_, 1, 16 * D_, 1, 6, 3);
    tdm_load_f32(lds_off(&Vs[0][0][0]), vbase, 16 * D_, 1, 16 * D_, 0, 0, 0);
  }

  for (int c = 0; c < NC; ++c) {
    const int cur = c & 1;
    if (w == 0) {
      if (c + 1 < NC) {
        tdm_load_f32(lds_off(&Ks[cur ^ 1][0][0]), kbase + (size_t)(c + 1) * 16 * D_,
                     16 * D_, 1, 16 * D_, 1, 6, 3);
        tdm_load_f32(lds_off(&Vs[cur ^ 1][0][0]), vbase + (size_t)(c + 1) * 16 * D_,
                     16 * D_, 1, 16 * D_, 0, 0, 0);
        __builtin_amdgcn_s_wait_tensorcnt(2);   // next chunk in flight, current done
      } else {
        __builtin_amdgcn_s_wait_tensorcnt(0);   // drain on last chunk
      }
    }
    __syncthreads();

    // scores: S = Q @ K^T   (B frag: B[k][n] = K[n][k])
    v8f sc = (v8f){};
#pragma unroll
    for (int ks = 0; ks < 32; ++ks) {
      const int ca = ks * 4 + (hi ? 2 : 0);
      v2f bf;
      bf.x = Ks[cur][lr][ca]; bf.y = Ks[cur][lr][ca + 1];
      sc = __builtin_amdgcn_wmma_f32_16x16x4_f32(false, qa[ks], false, bf, (short)0, sc, false, false);
    }

    // online softmax; row v lives in VGPR v across a 16-lane half-group
    float p[8];
#pragma unroll
    for (int v = 0; v < 8; ++v) {
      int sr = srow + v + (hi ? 8 : 0);
      float s = sc[v] * invscale + mbase[(size_t)sr * L_ + c * 16 + lr];
      float cm = s;
      cm = fmaxf(cm, __shfl_xor(cm, 1, 32));
      cm = fmaxf(cm, __shfl_xor(cm, 2, 32));
      cm = fmaxf(cm, __shfl_xor(cm, 4, 32));
      cm = fmaxf(cm, __shfl_xor(cm, 8, 32));
      float mn = fmaxf(m_run[v], cm);
      float f  = __expf(m_run[v] - mn);
      float pv = __expf(s - mn);
      float rs = pv;
      rs += __shfl_xor(rs, 1, 32);
      rs += __shfl_xor(rs, 2, 32);
      rs += __shfl_xor(rs, 4, 32);
      rs += __shfl_xor(rs, 8, 32);
      l_run[v] = l_run[v] * f + rs;
      m_run[v] = mn;
#pragma unroll
      for (int dt = 0; dt < 8; ++dt) acc[dt][v] *= f;
      p[v] = pv;
    }

    // transpose P (C/D layout -> A layout) through per-wave LDS scratch
#pragma unroll
    for (int v = 0; v < 8; ++v)
      Ps[w][v + (hi ? 8 : 0)][lr] = p[v];
    __builtin_amdgcn_wave_barrier();
    asm volatile("s_wait_dscnt 0" ::: "memory");
    __builtin_amdgcn_wave_barrier();

    // O += P @ V
#pragma unroll
    for (int ks = 0; ks < 4; ++ks) {
      const int ca = ks * 4 + (hi ? 2 : 0);
      v2f pa;
      pa.x = Ps[w][lr][ca]; pa.y = Ps[w][lr][ca + 1];
#pragma unroll
      for (int dt = 0; dt < 8; ++dt) {
        v2f bf;
        bf.x = Vs[cur][ca][dt * 16 + lr];
        bf.y = Vs[cur][ca + 1][dt * 16 + lr];
        acc[dt] = __builtin_amdgcn_wmma_f32_16x16x4_f32(false, pa, false, bf, (short)0, acc[dt], false, false);
      }
    }
    __syncthreads();
  }

  // normalize and store as (B,S,H*D)
  float* obase = attn + (size_t)b * S_ * HID_ + (size_t)h * D_;
#pragma unroll
  for (int v = 0; v < 8; ++v) {
    int sr = srow + v + (hi ? 8 : 0);
    float inv = 1.0f / l_run[v];
#pragma unroll
    for (int dt = 0; dt < 8; ++dt)
      obase[(size_t)sr * HID_ + dt * 16 + lr] = acc[dt][v] * inv;
  }
}

// ---------------------------------------------------------------------------
extern "C" void kernel_launch(void* const* d_in, const int* in_sizes, int n_in,
                              void* d_out, int out_size, void* d_ws, size_t ws_size,
                              hipStream_t stream) {
  const float* x    = (const float*)d_in[0];
  const float* mask = (const float*)d_in[1];
  const float* cosp = (const float*)d_in[2];
  const float* sinp = (const float*)d_in[3];
  const float* kc   = (const float*)d_in[4];
  const float* vc   = (const float*)d_in[5];
  const float* Wq   = (const float*)d_in[8];
  const float* bq   = (const float*)d_in[9];
  const float* Wk   = (const float*)d_in[10];
  const float* bk   = (const float*)d_in[11];
  const float* Wv   = (const float*)d_in[12];
  const float* bv   = (const float*)d_in[13];
  const float* Wo   = (const float*)d_in[14];
  const float* bo   = (const float*)d_in[15];

  float* out  = (float*)d_out;                     // (B,S,HID)
  float* kout = out + (size_t)BS_ * HID_;          // (B,S,H,D)
  float* vout = kout + (size_t)BS_ * HID_;         // (B,S,H,D)

  float* ws = (float*)d_ws;
  const size_t M4 = (size_t)BS_ * HID_;            // 4M floats
  float* q_raw = ws;
  float* k_raw = ws + M4;
  float* v_raw = ws + 2 * M4;
  float* q_ws  = ws + 3 * M4;                      // (B,H,S,D)
  float* kfull = ws + 4 * M4;                      // (B,H,L,D) = 2*M4
  float* vfull = ws + 6 * M4;                      // (B,H,L,D) = 2*M4
  float* attnb = ws;                               // reuses dead q/k/v_raw

  dim3 gg(HID_ / 64, BS_ / 64);
  gemm_bias_wmma<<<gg, 256, 0, stream>>>(x, Wq, bq, q_raw);
  gemm_bias_wmma<<<gg, 256, 0, stream>>>(x, Wk, bk, k_raw);
  gemm_bias_wmma<<<gg, 256, 0, stream>>>(x, Wv, bv, v_raw);
  rope_scatter<<<(B_ * S_ * H_ * 64) / 256, 256, 0, stream>>>(
      q_raw, k_raw, v_raw, cosp, sinp, q_ws, kfull, vfull, kout, vout);
  cache_copy<<<(B_ * H_ * POS_ * D_) / 256, 256, 0, stream>>>(kc, vc, kfull, vfull);
  flash_attn<<<dim3(B_ * H_, S_ / 64), 128, 0, stream>>>(q_ws, kfull, vfull, mask, attnb);
  gemm_bias_wmma<<<gg, 256, 0, stream>>>(attnb, Wo, bo, out);
}